// Bone_loss_6665789243642
// MI455X (gfx1250) — compile-verified
//
#include <hip/hip_runtime.h>
#include <math.h>

typedef __attribute__((ext_vector_type(2))) float v2f;
typedef __attribute__((ext_vector_type(8))) float v8f;

#define NJ 17
#define NBONE 12
#define HW 4096

// Stage 1: one thread per sample -> per-block partial sums (deterministic order).
__global__ __launch_bounds__(128)
void bone_partial_kernel(const float* __restrict__ output,
                         const float* __restrict__ mask,
                         const int*   __restrict__ ind,
                         const float* __restrict__ target,
                         const float* __restrict__ gt2d,
                         float* __restrict__ partial,
                         int B)
{
    const int   ID1[NBONE] = {3, 2, 4, 5, 16, 15, 11, 12, 1, 4, 14, 11};
    const int   ID2[NBONE] = {2, 1, 5, 6, 15, 14, 12, 13, 0, 0,  8,  8};
    const float Wb [NBONE] = {1.0085885098415446f, 1.f, 1.f, 1.0085885098415446f,
                              1.1375361376887123f, 1.f, 1.f, 1.1375361376887123f,
                              1.f, 1.f, 1.f, 1.f};
    const int   GID[NBONE] = {0, 0, 0, 0, 1, 1, 1, 1, 2, 2, 3, 3};

    const int b = blockIdx.x * blockDim.x + threadIdx.x;
    float per_sample = 0.0f;

    if (b < B) {
        const float* tb = target + (size_t)b * NJ;
        const float* mb = mask   + (size_t)b * NJ;
        const int*   ib = ind    + (size_t)b * NJ;
        const float* gb = gt2d   + (size_t)b * NJ * 2;
        const float* ob = output + (size_t)b * HW;

        // Issue all gathers first so the 17 random loads overlap (one LOADcnt batch).
        int idx[NJ];
        #pragma unroll
        for (int j = 0; j < NJ; ++j) idx[j] = ib[j];
        float pr[NJ];
        #pragma unroll
        for (int j = 0; j < NJ; ++j) pr[j] = ob[idx[j]];

        float tj[NJ], gx[NJ], gy[NJ], msum = 0.0f;
        #pragma unroll
        for (int j = 0; j < NJ; ++j) {
            tj[j] = tb[j];
            msum += mb[j];
            gx[j] = gb[2 * j + 0];
            gy[j] = gb[2 * j + 1];
        }

        float num[4]   = {0.f, 0.f, 0.f, 0.f};
        float sum_l[4] = {0.f, 0.f, 0.f, 0.f};
        float l[NBONE];
        bool  vis[NBONE];

        #pragma unroll
        for (int k = 0; k < NBONE; ++k) {
            const int a = ID1[k], c = ID2[k];
            const bool v = (tj[a] > 0.5f) && (tj[c] > 0.5f);
            const float dx = gx[a] - gx[c];
            const float dy = gy[a] - gy[c];
            const float dp = pr[a] - pr[c];
            const float d2 = dx * dx + dy * dy + dp * dp;
            const float lv = v ? sqrtf(d2) * Wb[k] : 0.0f;
            vis[k] = v;
            l[k]   = lv;
            if (v) { num[GID[k]] += 1.0f; sum_l[GID[k]] += lv; }
        }

        float E[4], nm[4];
        #pragma unroll
        for (int g = 0; g < 4; ++g) {
            nm[g] = fmaxf(num[g], 1.0f);
            E[g]  = sum_l[g] / nm[g];
        }

        #pragma unroll
        for (int k = 0; k < NBONE; ++k) {
            if (vis[k] && l[k] > 0.0f) {
                const float d = l[k] - E[GID[k]];
                per_sample += d * d / (2.0f * nm[GID[k]]);
            }
        }

        if (msum != 0.0f) per_sample = 0.0f;  // active = (mask.sum == 0)
    }

    // Deterministic in-block reduction: wave32 shuffle tree, then 4 wave results.
    float v = per_sample;
    #pragma unroll
    for (int off = 16; off > 0; off >>= 1)
        v += __shfl_down(v, off, 32);

    __shared__ float sdata[4];
    if ((threadIdx.x & 31) == 0) sdata[threadIdx.x >> 5] = v;
    __syncthreads();
    if (threadIdx.x == 0)
        partial[blockIdx.x] = (sdata[0] + sdata[1]) + (sdata[2] + sdata[3]);
}

// Stage 2: sum up to 64 partials with one V_WMMA_F32_16X16X4_F32.
// A (16x4) holds the 64 partials (any bijective packing is fine because the
// B matrix is all-ones: D[m][n] = sum_k A[m][k], so sum of the 16 row-sums is
// the grand total). D layout: VGPR r, lanes 0-15 -> M=r; lanes 16-31 -> M=8+r.
// Each lane sums its 8 D regs (= row-sums M=0..7 or M=8..15 of one column),
// a single shfl_xor(16) combines the halves.
__global__ void reduce_wmma_kernel(const float* __restrict__ partial,
                                   float* __restrict__ out,
                                   int count, float scale)
{
    const int lane = threadIdx.x;  // exactly one wave32, EXEC all ones

    v2f a;
    const int i0 = 2 * lane, i1 = 2 * lane + 1;
    a.x = (i0 < count) ? partial[i0] : 0.0f;
    a.y = (i1 < count) ? partial[i1] : 0.0f;

    v2f bm;
    bm.x = 1.0f;
    bm.y = 1.0f;

    v8f c = {};
    // (neg_a, A, neg_b, B, c_mod, C, reuse_a, reuse_b)
    v8f d = __builtin_amdgcn_wmma_f32_16x16x4_f32(
        false, a, false, bm, (short)0, c, false, false);

    float s = 0.0f;
    #pragma unroll
    for (int r = 0; r < 8; ++r) s += d[r];

    s += __shfl_xor(s, 16, 32);  // M=0..7 half + M=8..15 half -> grand total

    if (lane == 0) out[0] = s * scale;
}

extern "C" void kernel_launch(void* const* d_in, const int* in_sizes, int n_in,
                              void* d_out, int out_size, void* d_ws, size_t ws_size,
                              hipStream_t stream)
{
    const float* output = (const float*)d_in[0];  // [B,1,64,64]
    const float* mask   = (const float*)d_in[1];  // [B,17]
    const int*   ind    = (const int*)  d_in[2];  // [B,17]
    const float* target = (const float*)d_in[3];  // [B,17]
    const float* gt2d   = (const float*)d_in[4];  // [B,17,2]

    const int B = in_sizes[1] / NJ;               // 8192
    float* partial = (float*)d_ws;

    const int threads = 128;
    const int blocks  = (B + threads - 1) / threads;  // 64 for B=8192

    bone_partial_kernel<<<blocks, threads, 0, stream>>>(
        output, mask, ind, target, gt2d, partial, B);

    const float scale = 1.0f / (float)B;          // _VAR_WEIGHT == 1.0
    reduce_wmma_kernel<<<1, 32, 0, stream>>>(
        partial, (float*)d_out, blocks, scale);
}